// ProbAttention_80195629351289
// MI455X (gfx1250) — compile-verified
//
#include <hip/hip_runtime.h>
#include <math.h>

// Fixed problem shape from the reference setup_inputs()
#define B_  2
#define L_  2048
#define H_  16
#define D_  128
#define SK_ 40      // sample_k
#define NT_ 40      // n_top
#define MT_ 3       // ceil(NT_/16) query tiles of 16
#define SCALE_ 0.08838834764831845f   // 1/sqrt(128)

typedef float v2f __attribute__((ext_vector_type(2)));
typedef float v8f __attribute__((ext_vector_type(8)));

// ---------------------------------------------------------------------------
// K1: vmean[b,h,d] = mean_l V[b,l,h,d]       (input layout is B,L,H,D)
// ---------------------------------------------------------------------------
__global__ __launch_bounds__(128) void vmean_kernel(const float* __restrict__ V,
                                                    float* __restrict__ vmean) {
  const int bh = blockIdx.x;
  const int b = bh / H_, h = bh % H_;
  const int d = threadIdx.x;
  float s = 0.f;
  for (int l = 0; l < L_; ++l)
    s += V[((size_t)(b * L_ + l) * H_ + h) * D_ + d];
  vmean[(b * H_ + h) * D_ + d] = s * (1.0f / L_);
}

// ---------------------------------------------------------------------------
// K2: sparsity measure  M[b,h,q] = max_s(Q.Ks) - sum_s(Q.Ks)/L
// One wave32 per (b,h,q); Q row held as float4 per lane; wave reduction.
// ---------------------------------------------------------------------------
__global__ __launch_bounds__(256) void sparsity_kernel(const float* __restrict__ Q,
                                                       const float* __restrict__ K,
                                                       const int* __restrict__ idx,
                                                       float* __restrict__ M) {
  const int wid  = blockIdx.x * 8 + (threadIdx.x >> 5);
  const int lane = threadIdx.x & 31;
  const int b = wid / (H_ * L_);
  const int rem = wid % (H_ * L_);
  const int h = rem / L_, q = rem % L_;

  const float4 qv = ((const float4*)&Q[((size_t)(b * L_ + q) * H_ + h) * D_])[lane];
  float mx = -__builtin_inff(), sm = 0.f;
  for (int s = 0; s < SK_; ++s) {
    const int kq = idx[q * SK_ + s];
    const float4 kv = ((const float4*)&K[((size_t)(b * L_ + kq) * H_ + h) * D_])[lane];
    float p = qv.x * kv.x + qv.y * kv.y + qv.z * kv.z + qv.w * kv.w;
    #pragma unroll
    for (int m = 16; m >= 1; m >>= 1) p += __shfl_xor(p, m, 32);
    mx = fmaxf(mx, p);
    sm += p;
  }
  if (lane == 0) M[(b * H_ + h) * L_ + q] = mx - sm * (1.0f / L_);
}

// ---------------------------------------------------------------------------
// K3: top-40 indices of M per (b,h) by iterative argmax in LDS.
// Order of indices is irrelevant (result is scattered back by index).
// ---------------------------------------------------------------------------
__global__ __launch_bounds__(256) void topk_kernel(const float* __restrict__ M,
                                                   int* __restrict__ mtop) {
  __shared__ float sv[L_];
  __shared__ float rv[256];
  __shared__ int   ri[256];
  const int bh = blockIdx.x;
  const int tid = threadIdx.x;
  for (int i = tid; i < L_; i += 256) sv[i] = M[(size_t)bh * L_ + i];
  __syncthreads();
  for (int t = 0; t < NT_; ++t) {
    float bv = -__builtin_inff(); int bi = -1;
    for (int i = tid; i < L_; i += 256) {
      const float v = sv[i];
      if (v > bv || (v == bv && i < bi)) { bv = v; bi = i; }
    }
    rv[tid] = bv; ri[tid] = bi;
    __syncthreads();
    for (int s = 128; s > 0; s >>= 1) {
      if (tid < s) {
        if (rv[tid + s] > rv[tid] ||
            (rv[tid + s] == rv[tid] && ri[tid + s] < ri[tid])) {
          rv[tid] = rv[tid + s]; ri[tid] = ri[tid + s];
        }
      }
      __syncthreads();
    }
    if (tid == 0) { mtop[bh * NT_ + t] = ri[0]; sv[ri[0]] = -__builtin_inff(); }
    __syncthreads();
  }
}

// ---------------------------------------------------------------------------
// K4: out[b,l,h,d] = vmean[b,h,d]  (baseline context; also inits poisoned d_out)
// ---------------------------------------------------------------------------
__global__ __launch_bounds__(256) void fill_kernel(const float* __restrict__ vmean,
                                                   float* __restrict__ out) {
  const size_t i = (size_t)blockIdx.x * 256 + threadIdx.x;
  const int d = (int)(i % D_);
  const int h = (int)((i / D_) % H_);
  const int b = (int)(i / ((size_t)D_ * H_ * L_));
  out[i] = vmean[(b * H_ + h) * D_ + d];
}

// ---------------------------------------------------------------------------
// K5: flash attention for the selected queries, fp32 WMMA 16x16x4.
// One wave32 per (b, h, query-tile).  A-frag element offset = 4*kk + 2*half
// (ISA 7.12.2 32-bit A 16x4 layout); C/D row = v + 8*half, col = lane&15.
// ---------------------------------------------------------------------------
__global__ __launch_bounds__(32) void attn_kernel(const float* __restrict__ Q,
                                                  const float* __restrict__ K,
                                                  const float* __restrict__ V,
                                                  const int* __restrict__ mtop,
                                                  float* __restrict__ out) {
  const int bid = blockIdx.x;
  const int mt = bid % MT_;
  const int h  = (bid / MT_) % H_;
  const int b  = bid / (MT_ * H_);
  const int lane = threadIdx.x;
  const int half = lane >> 4;
  const int r = lane & 15;

  __shared__ float kt[16 * D_];
  __shared__ float vt[16 * D_];
  __shared__ float pls[16 * 16];

  // Q rows (A fragments) for this tile; rows beyond n_top are zero-padded.
  v2f qa[32];
  const int u = mt * 16 + r;
  const int qrow = (u < NT_) ? mtop[(b * H_ + h) * NT_ + u] : -1;
  if (qrow >= 0) {
    const float* qp = &Q[((size_t)(b * L_ + qrow) * H_ + h) * D_];
    #pragma unroll
    for (int kk = 0; kk < 32; ++kk) {
      const float2 t2 = *(const float2*)&qp[4 * kk + 2 * half];
      v2f a; a[0] = t2.x; a[1] = t2.y;
      qa[kk] = a;
    }
  } else {
    #pragma unroll
    for (int kk = 0; kk < 32; ++kk) { v2f a; a[0] = 0.f; a[1] = 0.f; qa[kk] = a; }
  }

  v8f o[8];
  #pragma unroll
  for (int i = 0; i < 8; ++i) o[i] = {};
  float m_run[8], l_run[8];
  #pragma unroll
  for (int v = 0; v < 8; ++v) { m_run[v] = -__builtin_inff(); l_run[v] = 0.f; }

  for (int t = 0; t < L_ / 16; ++t) {
    const int n0 = t * 16;
    __syncthreads();  // protect kt/vt/pls from previous iteration's readers
    // Stage K and V tiles (16 x 128 fp32 each) into LDS, coalesced float4.
    for (int rr = half; rr < 16; rr += 2) {
      const float* kp = &K[((size_t)(b * L_ + n0 + rr) * H_ + h) * D_];
      const float* vp = &V[((size_t)(b * L_ + n0 + rr) * H_ + h) * D_];
      *(float4*)&kt[rr * D_ + r * 8 + 0] = *(const float4*)&kp[r * 8 + 0];
      *(float4*)&kt[rr * D_ + r * 8 + 4] = *(const float4*)&kp[r * 8 + 4];
      *(float4*)&vt[rr * D_ + r * 8 + 0] = *(const float4*)&vp[r * 8 + 0];
      *(float4*)&vt[rr * D_ + r * 8 + 4] = *(const float4*)&vp[r * 8 + 4];
    }
    __syncthreads();

    if (t + 1 < L_ / 16) {  // gfx1250 global_prefetch_b8 hint for next tile
      __builtin_prefetch(&K[((size_t)(b * L_ + n0 + 16 + r) * H_ + h) * D_], 0, 1);
      __builtin_prefetch(&V[((size_t)(b * L_ + n0 + 16 + r) * H_ + h) * D_], 0, 1);
    }

    // S(16x16) = Q_tile(16x128) x K_tile^T(128x16), 32 fp32 WMMA k-steps.
    v8f s = {};
    #pragma unroll
    for (int kk = 0; kk < 32; ++kk) {
      const float2 k2 = *(const float2*)&kt[r * D_ + 4 * kk + 2 * half];
      v2f kb; kb[0] = k2.x; kb[1] = k2.y;
      s = __builtin_amdgcn_wmma_f32_16x16x4_f32(false, qa[kk], false, kb,
                                                (short)0, s, false, false);
    }

    // Online softmax per row (rows live in 16-lane halves; xor masks 1..8).
    float corr[8];
    #pragma unroll
    for (int v = 0; v < 8; ++v) {
      const float x = s[v] * SCALE_;
      float rm = x;
      #pragma unroll
      for (int msk = 8; msk >= 1; msk >>= 1) rm = fmaxf(rm, __shfl_xor(rm, msk, 32));
      const float mnew = fmaxf(m_run[v], rm);
      const float p = __expf(x - mnew);
      float rs = p;
      #pragma unroll
      for (int msk = 8; msk >= 1; msk >>= 1) rs += __shfl_xor(rs, msk, 32);
      const float c = __expf(m_run[v] - mnew);
      l_run[v] = l_run[v] * c + rs;
      m_run[v] = mnew;
      corr[v] = c;
      s[v] = p;
    }

    // Stage P through LDS to convert C-layout -> A-layout; rescale O.
    #pragma unroll
    for (int v = 0; v < 8; ++v) pls[(v + 8 * half) * 16 + r] = s[v];
    #pragma unroll
    for (int ds = 0; ds < 8; ++ds) {
      #pragma unroll
      for (int v = 0; v < 8; ++v) o[ds][v] *= corr[v];
    }
    __syncthreads();

    v2f pa[4];
    #pragma unroll
    for (int kk = 0; kk < 4; ++kk) {
      v2f a;
      a[0] = pls[r * 16 + 4 * kk + 2 * half];
      a[1] = pls[r * 16 + 4 * kk + 2 * half + 1];
      pa[kk] = a;
    }

    // O(16x128) += P(16x16) x V_tile(16x128): 8 d-subtiles x 4 k-steps.
    #pragma unroll
    for (int ds = 0; ds < 8; ++ds) {
      const int d0 = ds * 16;
      v8f acc = o[ds];
      #pragma unroll
      for (int kk = 0; kk < 4; ++kk) {
        v2f vb;
        vb[0] = vt[(4 * kk + 2 * half) * D_ + d0 + r];
        vb[1] = vt[(4 * kk + 2 * half + 1) * D_ + d0 + r];
        acc = __builtin_amdgcn_wmma_f32_16x16x4_f32(false, pa[kk], false, vb,
                                                    (short)0, acc, false, false);
      }
      o[ds] = acc;
    }
  }

  // Epilogue: normalize and scatter rows back to their query positions.
  #pragma unroll
  for (int v = 0; v < 8; ++v) {
    const int m = v + 8 * half;
    const int uu = mt * 16 + m;
    if (uu < NT_) {
      const int qd = mtop[(b * H_ + h) * NT_ + uu];
      const float inv = 1.0f / l_run[v];
      #pragma unroll
      for (int ds = 0; ds < 8; ++ds)
        out[((size_t)(b * L_ + qd) * H_ + h) * D_ + ds * 16 + r] = o[ds][v] * inv;
    }
  }
}

// ---------------------------------------------------------------------------
extern "C" void kernel_launch(void* const* d_in, const int* in_sizes, int n_in,
                              void* d_out, int out_size, void* d_ws, size_t ws_size,
                              hipStream_t stream) {
  const float* Q   = (const float*)d_in[0];
  const float* K   = (const float*)d_in[1];
  const float* V   = (const float*)d_in[2];
  const int*   idx = (const int*)d_in[3];
  float* out = (float*)d_out;

  // Workspace layout (~280 KB total)
  float* vmean = (float*)d_ws;                  // B*H*D
  float* M     = vmean + B_ * H_ * D_;          // B*H*L
  int*   mtop  = (int*)(M + B_ * H_ * L_);      // B*H*NT

  vmean_kernel<<<B_ * H_, 128, 0, stream>>>(V, vmean);
  sparsity_kernel<<<(B_ * H_ * L_) / 8, 256, 0, stream>>>(Q, K, idx, M);
  topk_kernel<<<B_ * H_, 256, 0, stream>>>(M, mtop);
  fill_kernel<<<(B_ * L_ * H_ * D_) / 256, 256, 0, stream>>>(vmean, out);
  attn_kernel<<<B_ * H_ * MT_, 32, 0, stream>>>(Q, K, V, mtop, out);
}